// ReconstructionModel_74234214744253
// MI455X (gfx1250) — compile-verified
//
#include <hip/hip_runtime.h>
#include <hip/hip_bf16.h>

typedef __attribute__((ext_vector_type(2))) float v2f;
typedef __attribute__((ext_vector_type(8))) float v8f;

#define NB   16
#define FEAT 161
#define SIDE 48
#define MLEN 32

// ---------------- packed parameter layout (float offsets in d_ws) ----------------
static __host__ __device__ inline int KS(int t)   { const int k[6] = {25,21,17,13,9,5}; return k[t]; }
static __host__ __device__ inline int CIN(int i)  { const int c[4] = {161,80,40,20};    return c[i]; }
static __host__ __device__ inline int COUT(int i) { const int c[4] = {80,40,20,10};     return c[i]; }
static __host__ __device__ inline int CP(int i)   { const int c[4] = {176,80,48,32};    return c[i]; }  // round16(Cin)
static __host__ __device__ inline int CO16(int i) { const int c[4] = {80,48,32,16};     return c[i]; }  // round16(Cout)

// per tower: padded t-major weights [CO16][ks*CP] per layer, then b0..b3, a0..a3
// layer W row-blocks: 80*176, 48*80, 32*48, 16*32 (x ks) => 19968*ks total
static __host__ __device__ inline int towerBase(int t){
  int o = 0; for (int j = 0; j < t; j++) o += 19968*KS(j) + 170; return o;
}
static __host__ __device__ inline int twrW(int t,int i){
  const int c[4] = {0, 14080, 17920, 19456}; return towerBase(t) + c[i]*KS(t);
}
static __host__ __device__ inline int twrB(int t,int i){
  const int c[4] = {0,80,120,140}; return towerBase(t) + 19968*KS(t) + c[i];
}
static __host__ __device__ inline int twrA(int t,int i){
  return towerBase(t) + 19968*KS(t) + 150 + 5*i;
}
#define DOWN_OFF 1798140            // 19968*90 + 6*170
static __host__ __device__ inline int downW(int s,int i){ return DOWN_OFF + s*1075 + 200*i; }
static __host__ __device__ inline int downB(int s,int i){ return DOWN_OFF + s*1075 + 1000 + 10*i; }
static __host__ __device__ inline int downA(int s,int i){ return DOWN_OFF + s*1075 + 1050 + 5*i; }
#define LIN_OFF 1800290             // DOWN_OFF + 2*1075
static __host__ __device__ inline int linW1(int s){ return LIN_OFF + s*14963; }
static __host__ __device__ inline int linB1(int s){ return linW1(s) + 860; }
static __host__ __device__ inline int linA1(int s){ return linW1(s) + 946; }
static __host__ __device__ inline int linW2(int s){ return linW1(s) + 951; }
static __host__ __device__ inline int linB2(int s){ return linW1(s) + 14797; }
static __host__ __device__ inline int linA2(int s){ return linW1(s) + 14958; }
#define FIN_OFF  1830216            // LIN_OFF + 2*14963
#define FIN_W1T  (FIN_OFF)          // transposed+padded [256][336] = 86016
#define FIN_W2T  (FIN_OFF + 86016)  // transposed+padded [176][256] = 45056
#define FIN_B1   (FIN_OFF + 131072)
#define FIN_A1   (FIN_B1 + 241)
#define FIN_B2   (FIN_A1 + 5)
#define FIN_A2   (FIN_B2 + 161)
#define PARAMS_END (FIN_A2 + 5)     // 1961700 (mult of 4)
// ---------------- activation workspace ----------------
#define INPW_OFF  1961700                        // [16][136][161]
#define LHIST_OFF (INPW_OFF + 16*136*161)        // [16][64][161]
#define RHIST_OFF (LHIST_OFF + 16*64*161)        // [16][64][161]
#define SRES_OFF  (RHIST_OFF + 16*64*161)        // [4][16][161]
// ---------------- step-kernel dynamic LDS layout (float offsets) ----------------
#define SM_XPAD 0                    // [72][176]
#define SM_BUF0 12672                // [72][80]
#define SM_BUF1 18432                // [72][48]
#define SM_BUF2 21888                // [72][32]
#define SM_AE   24192                // [72][12]
#define SM_DWN0 25056                // [10][48]
#define SM_DWN1 25536                // [10][48]
#define SM_MVEC 26016                // [10]
#define SM_H1   26026                // [86]
#define SM_TOTAL 26112               // floats -> 104448 bytes

__device__ __forceinline__ float taylor5(float x, float c0, float c1, float c2, float c3, float c4){
  return fmaf(x, fmaf(x, fmaf(x, fmaf(x, c4, c3), c2), c1), c0);
}

// Guard-free implicit-im2col conv + bias + Taylor, fp32 WMMA 16x16x4.
// in : LDS position-major [72][Cpad], 12-row zero halo, zero col-padding.
// Wp : global [Co16][ks*Cpad] t-major, zero-padded.  Kp multiple of 16 -> no remainder.
__device__ __forceinline__ void conv_wmma(const float* __restrict__ Wp,
                                          const float* __restrict__ bias,
                                          const float* __restrict__ tay,
                                          const float* in, int Cpad,
                                          float* out, int outStride,
                                          int Cout, int Co16, int ks, bool accum)
{
  const int Kp   = ks * Cpad;
  const int pad  = ks >> 1;
  const int lane = threadIdx.x & 31;
  const int wave = threadIdx.x >> 5;       // 8 waves
  const int row  = lane & 15;
  const int hi   = lane >> 4;
  const int tiles = (Co16 >> 4) * 3;       // N = 48 = 3 exact tiles
  const float c0 = tay[0], c1 = tay[1], c2 = tay[2], c3 = tay[3], c4 = tay[4];

  for (int tile = wave; tile < tiles; tile += 8) {
    const int mBase = (tile / 3) << 4;
    const int nBase = (tile % 3) << 4;
    const float* wrow = Wp + (size_t)(mBase + row) * Kp + 2*hi;
    // linear im2col: X[kk][n] = in[(n - pad + 12)*Cpad + kk]
    const float* brow = in + (size_t)(nBase + row - pad + 12) * Cpad + 2*hi;
    v8f acc = {0.f,0.f,0.f,0.f,0.f,0.f,0.f,0.f};
    #pragma unroll 4
    for (int k0 = 0; k0 < Kp; k0 += 4) {
      const v2f a  = *(const v2f*)(wrow + k0);
      const v2f bv = *(const v2f*)(brow + k0);
      acc = __builtin_amdgcn_wmma_f32_16x16x4_f32(false, a, false, bv, (short)0, acc, false, false);
    }
    #pragma unroll
    for (int r = 0; r < 8; r++) {
      const int om = mBase + r + (hi << 3);
      const int on = nBase + row;
      if (om < Cout) {
        const float y = taylor5(acc[r] + bias[om], c0, c1, c2, c3, c4);
        if (accum) out[(on + 12) * outStride + om] += y;
        else       out[(on + 12) * outStride + om] = y;
      }
    }
  }
}

// ---- repack kernels (read d_in leaves directly, every call) ----
__global__ void repack_conv(const float* __restrict__ src, float* __restrict__ dst,
                            int Cout, int Cin, int ks, int Cpad, int Co16)
{
  const int rowLen = ks * Cpad;
  const int total  = Co16 * rowLen;
  for (int idx = blockIdx.x * blockDim.x + threadIdx.x; idx < total; idx += gridDim.x * blockDim.x) {
    const int m = idx / rowLen;
    const int r = idx % rowLen;
    const int t = r / Cpad, ci = r % Cpad;
    dst[idx] = (m < Cout && ci < Cin) ? src[(m * Cin + ci) * ks + t] : 0.f;
  }
}

__global__ void repack_T(const float* __restrict__ src, float* __restrict__ dst,
                         int K, int N, int Kp, int Np)
{
  const int total = Np * Kp;
  for (int idx = blockIdx.x * blockDim.x + threadIdx.x; idx < total; idx += gridDim.x * blockDim.x) {
    const int n = idx / Kp, k = idx % Kp;
    dst[idx] = (k < K && n < N) ? src[k * N + n] : 0.f;
  }
}

__global__ void init_kernel(const float* __restrict__ x, float* __restrict__ inpw,
                            float* __restrict__ lh, float* __restrict__ rh)
{
  const int total = NB * 136 * FEAT;
  for (int idx = blockIdx.x * blockDim.x + threadIdx.x; idx < total; idx += gridDim.x * blockDim.x) {
    const int b = idx / (136 * FEAT);
    const int r = idx % (136 * FEAT);
    const int f = r / FEAT, c = r % FEAT;
    inpw[idx] = x[((size_t)b * 8192 + 200 + f) * FEAT + c];
    if (f < SIDE) {
      lh[((size_t)b * 64 + f)      * FEAT + c] = x[((size_t)b * 8192 + 207 + f) * FEAT + c];
      rh[((size_t)b * 64 + 16 + f) * FEAT + c] = x[((size_t)b * 8192 + 288 + f) * FEAT + c];
    }
  }
}

// One block per (step s, batch b): full forward_step fused in LDS.
__global__ __launch_bounds__(256) void step_kernel(
    const float* __restrict__ p, const float* __restrict__ inpw,
    const float* __restrict__ lh, const float* __restrict__ rh,
    float* __restrict__ sres, int l)
{
  extern __shared__ float smem[];
  float* xpad = smem + SM_XPAD;
  float* buf0 = smem + SM_BUF0;
  float* buf1 = smem + SM_BUF1;
  float* buf2 = smem + SM_BUF2;
  float* ae   = smem + SM_AE;
  float* dwn0 = smem + SM_DWN0;
  float* dwn1 = smem + SM_DWN1;
  float* mvec = smem + SM_MVEC;
  float* h1   = smem + SM_H1;

  const int tid  = threadIdx.x;
  const int s    = blockIdx.x >> 4;     // 0:lo 1:rlo 2:ro 3:lro
  const int b    = blockIdx.x & 15;
  const int side = (s == 0 || s == 3) ? 0 : 1;

  const float* src;
  if (s == 0)      src = lh   + (size_t)(b * 64  + l)      * FEAT;
  else if (s == 1) src = inpw + (size_t)(b * 136 + 56 + l) * FEAT;
  else if (s == 2) src = rh   + (size_t)(b * 64  + 16 - l) * FEAT;
  else             src = inpw + (size_t)(b * 136 + 40 + l) * FEAT;

  // zero inter-layer buffers (halos/col-pads must be zero); buf0..buf2+ae contiguous
  for (int idx = tid; idx < SM_DWN0 - SM_BUF0; idx += 256) buf0[idx] = 0.f;
  // fill xpad [72][176]: rows 12..59 = window, halo/col-pad zero
  for (int idx = tid; idx < 72 * 176; idx += 256) {
    const int r2 = idx / 176, c = idx % 176;
    const int j = r2 - 12;
    xpad[idx] = (j >= 0 && j < SIDE && c < FEAT) ? src[j * FEAT + c] : 0.f;
  }
  __syncthreads();

  for (int t = 0; t < 6; t++) {
    const int ks = KS(t);
    conv_wmma(p + twrW(t,0), p + twrB(t,0), p + twrA(t,0), xpad, 176, buf0, 80, 80, 80, ks, false);
    __syncthreads();
    conv_wmma(p + twrW(t,1), p + twrB(t,1), p + twrA(t,1), buf0,  80, buf1, 48, 40, 48, ks, false);
    __syncthreads();
    conv_wmma(p + twrW(t,2), p + twrB(t,2), p + twrA(t,2), buf1,  48, buf2, 32, 20, 32, ks, false);
    __syncthreads();
    conv_wmma(p + twrW(t,3), p + twrB(t,3), p + twrA(t,3), buf2,  32, ae,   12, 10, 16, ks, t != 0);
    __syncthreads();
  }

  // down tower: 5 x (k=2, dilation 2^i, valid) ; 48->47->45->41->33->17
  const int Lin0[5] = {48, 47, 45, 41, 33};
  for (int i = 0; i < 5; i++) {
    const int dil  = 1 << i;
    const int Lout = Lin0[i] - dil;
    const float* W    = p + downW(side, i);
    const float* bias = p + downB(side, i);
    const float* tay  = p + downA(side, i);
    float* dout = (i & 1) ? dwn1 : dwn0;
    const float* din = (i & 1) ? dwn0 : dwn1;   // valid for i>=1
    for (int idx = tid; idx < 10 * Lout; idx += 256) {
      const int co = idx / Lout, ll = idx % Lout;
      float sum = bias[co];
      #pragma unroll
      for (int ci = 0; ci < 10; ci++) {
        const float x0 = (i == 0) ? ae[(ll + 12) * 12 + ci]       : din[ci * 48 + ll];
        const float x1 = (i == 0) ? ae[(ll + dil + 12) * 12 + ci] : din[ci * 48 + ll + dil];
        sum = fmaf(W[(co * 10 + ci) * 2 + 0], x0, sum);
        sum = fmaf(W[(co * 10 + ci) * 2 + 1], x1, sum);
      }
      dout[co * 48 + ll] = taylor5(sum, tay[0], tay[1], tay[2], tay[3], tay[4]);
    }
    __syncthreads();
  }
  // final down output (i=4) is in dwn0, length 17
  if (tid < 10) {
    float sum = 0.f;
    for (int j = 0; j < 17; j++) sum += dwn0[tid * 48 + j];
    mvec[tid] = sum * (1.f / 17.f);
  }
  __syncthreads();
  { // lin1: 10 -> 86
    const float* W = p + linW1(side); const float* bias = p + linB1(side); const float* tay = p + linA1(side);
    if (tid < 86) {
      float sum = bias[tid];
      #pragma unroll
      for (int ci = 0; ci < 10; ci++) sum = fmaf(mvec[ci], W[ci * 86 + tid], sum);
      h1[tid] = taylor5(sum, tay[0], tay[1], tay[2], tay[3], tay[4]);
    }
  }
  __syncthreads();
  { // lin2: 86 -> 161
    const float* W = p + linW2(side); const float* bias = p + linB2(side); const float* tay = p + linA2(side);
    if (tid < FEAT) {
      float sum = bias[tid];
      for (int j = 0; j < 86; j++) sum = fmaf(h1[j], W[j * FEAT + tid], sum);
      sres[((size_t)s * NB + b) * FEAT + tid] = taylor5(sum, tay[0], tay[1], tay[2], tay[3], tay[4]);
    }
  }
}

// final MLP (concat 2x161 -> 241 -> 161, clamp[0,6]) via WMMA + state scatter.
__global__ __launch_bounds__(512) void final_kernel(
    const float* __restrict__ p, const float* __restrict__ sres,
    float* __restrict__ inpw, float* __restrict__ lh, float* __restrict__ rh,
    float* __restrict__ out, int l)
{
  __shared__ float h [16 * 340];   // K padded to 336 (loop bound), stride 340 (bank spread)
  __shared__ float h2[16 * 260];   // K padded to 256 (loop bound), stride 260
  const int o   = blockIdx.x;      // 0: (lo,rlo)->pos l ; 1: (ro,lro)->pos 31-l
  const int tid = threadIdx.x;
  const float* sA = sres + (size_t)(o ? 2 : 0) * NB * FEAT;
  const float* sB = sA + NB * FEAT;

  for (int idx = tid; idx < 16 * 340; idx += 512) {
    const int bb = idx / 340, c = idx % 340;
    float v = 0.f;
    if (c < FEAT)          v = sA[bb * FEAT + c];
    else if (c < 2 * FEAT) v = sB[bb * FEAT + (c - FEAT)];
    h[idx] = v;
  }
  for (int idx = tid; idx < 16 * 260; idx += 512) h2[idx] = 0.f;
  __syncthreads();

  const int lane = tid & 31, wave = tid >> 5;
  const int row = lane & 15, hi = lane >> 4;

  { // layer 1: [16 x 322] @ [322 x 241] ; Wt1 [256][336]
    const float* Wt   = p + FIN_W1T;
    const float* bias = p + FIN_B1; const float* tay = p + FIN_A1;
    const int nBase = wave << 4;
    const int n = nBase + row;
    const float* hrow = h  + row * 340 + 2*hi;
    const float* wrow = Wt + (size_t)n * 336 + 2*hi;
    v8f acc = {0.f,0.f,0.f,0.f,0.f,0.f,0.f,0.f};
    #pragma unroll 4
    for (int k0 = 0; k0 < 336; k0 += 4) {
      const v2f a  = *(const v2f*)(hrow + k0);
      const v2f bv = *(const v2f*)(wrow + k0);
      acc = __builtin_amdgcn_wmma_f32_16x16x4_f32(false, a, false, bv, (short)0, acc, false, false);
    }
    #pragma unroll
    for (int r = 0; r < 8; r++) {
      const int ob = r + (hi << 3), on = nBase + row;
      if (on < 241) h2[ob * 260 + on] = taylor5(acc[r] + bias[on], tay[0], tay[1], tay[2], tay[3], tay[4]);
    }
  }
  __syncthreads();

  const int pos = o ? (31 - l) : l;
  if (wave < 11) { // layer 2: [16 x 241] @ [241 x 161] ; Wt2 [176][256]
    const float* Wt   = p + FIN_W2T;
    const float* bias = p + FIN_B2; const float* tay = p + FIN_A2;
    const int nBase = wave << 4;
    const int n = nBase + row;
    const float* hrow = h2 + row * 260 + 2*hi;
    const float* wrow = Wt + (size_t)n * 256 + 2*hi;
    v8f acc = {0.f,0.f,0.f,0.f,0.f,0.f,0.f,0.f};
    #pragma unroll 4
    for (int k0 = 0; k0 < 256; k0 += 4) {
      const v2f a  = *(const v2f*)(hrow + k0);
      const v2f bv = *(const v2f*)(wrow + k0);
      acc = __builtin_amdgcn_wmma_f32_16x16x4_f32(false, a, false, bv, (short)0, acc, false, false);
    }
    #pragma unroll
    for (int r = 0; r < 8; r++) {
      const int ob = r + (hi << 3), on = nBase + row;
      if (on < FEAT) {
        float v = taylor5(acc[r] + bias[on], tay[0], tay[1], tay[2], tay[3], tay[4]);
        v = fminf(fmaxf(v, 0.f), 6.f);
        out [((size_t)ob * MLEN + pos)      * FEAT + on] = v;
        inpw[((size_t)ob * 136  + 56 + pos) * FEAT + on] = v;
      }
    }
  }

  if (o == 0) { // lo -> left_hist slot 48+l
    for (int idx = tid; idx < NB * FEAT; idx += 512) {
      const int bb = idx / FEAT, c = idx % FEAT;
      lh[((size_t)bb * 64 + 48 + l) * FEAT + c] = sA[idx];
    }
    if (l == 0) { // first scatter: rest of masked region becomes zeros
      for (int idx = tid; idx < NB * 30 * FEAT; idx += 512) {
        const int bb = idx / (30 * FEAT), r2 = idx % (30 * FEAT);
        const int f = r2 / FEAT + 1, c = r2 % FEAT;
        inpw[((size_t)bb * 136 + 56 + f) * FEAT + c] = 0.f;
      }
    }
  } else {      // ro -> right_hist slot 15-l
    for (int idx = tid; idx < NB * FEAT; idx += 512) {
      const int bb = idx / FEAT, c = idx % FEAT;
      rh[((size_t)bb * 64 + 15 - l) * FEAT + c] = sA[idx];
    }
  }
}

extern "C" void kernel_launch(void* const* d_in, const int* in_sizes, int n_in,
                              void* d_out, int out_size, void* d_ws, size_t ws_size,
                              hipStream_t stream) {
  (void)in_sizes; (void)n_in; (void)out_size; (void)ws_size;
  const float* x = (const float*)d_in[0];
  float* p    = (float*)d_ws;
  float* inpw = p + INPW_OFF;
  float* lh   = p + LHIST_OFF;
  float* rh   = p + RHIST_OFF;
  float* sres = p + SRES_OFF;
  float* out  = (float*)d_out;

  // Consume pytree leaves in order (sorted dict keys: down, final, lin, towers).
  int idx = 2;  // d_in[0]=x, d_in[1]=mask
  auto cp = [&](int i, long long dstF, long long nF) {
    hipMemcpyAsync(p + dstF, d_in[i], (size_t)nF * sizeof(float), hipMemcpyDeviceToDevice, stream);
  };
  for (int s = 0; s < 2; s++) {                          // down: a,b,w
    for (int i = 0; i < 5; i++) cp(idx++, downA(s, i), 5);
    for (int i = 0; i < 5; i++) cp(idx++, downB(s, i), 10);
    for (int i = 0; i < 5; i++) cp(idx++, downW(s, i), 200);
  }
  cp(idx++, FIN_A1, 5);   cp(idx++, FIN_A2, 5);          // final: a,b,w
  cp(idx++, FIN_B1, 241); cp(idx++, FIN_B2, 161);
  hipLaunchKernelGGL(repack_T, dim3(256), dim3(256), 0, stream,
                     (const float*)d_in[idx++], p + FIN_W1T, 322, 241, 336, 256);
  hipLaunchKernelGGL(repack_T, dim3(256), dim3(256), 0, stream,
                     (const float*)d_in[idx++], p + FIN_W2T, 241, 161, 256, 176);
  for (int s = 0; s < 2; s++) {                          // lin: a,b,w
    cp(idx++, linA1(s), 5);   cp(idx++, linA2(s), 5);
    cp(idx++, linB1(s), 86);  cp(idx++, linB2(s), 161);
    cp(idx++, linW1(s), 860); cp(idx++, linW2(s), 13846);
  }
  for (int t = 0; t < 6; t++) {                          // towers: a,b,w
    for (int i = 0; i < 4; i++) cp(idx++, twrA(t, i), 5);
    for (int i = 0; i < 4; i++) cp(idx++, twrB(t, i), COUT(i));
    for (int i = 0; i < 4; i++)
      hipLaunchKernelGGL(repack_conv, dim3(256), dim3(256), 0, stream,
                         (const float*)d_in[idx++], p + twrW(t, i),
                         COUT(i), CIN(i), KS(t), CP(i), CO16(i));
  }

  hipLaunchKernelGGL(init_kernel, dim3(256), dim3(256), 0, stream, x, inpw, lh, rh);
  for (int l = 0; l < MLEN / 2; l++) {
    hipLaunchKernelGGL(step_kernel,  dim3(64), dim3(256), SM_TOTAL * sizeof(float), stream,
                       p, inpw, lh, rh, sres, l);
    hipLaunchKernelGGL(final_kernel, dim3(2),  dim3(512), 0, stream,
                       p, sres, inpw, lh, rh, out, l);
  }
}